// Net_69655779607086
// MI455X (gfx1250) — compile-verified
//
#include <hip/hip_runtime.h>

#define NN 100000
#define RR 50
#define BB 30
#define HH 16
#define CC 16
#define EE 1000000
#define NTILES (NN / 16)   // 6250

typedef __attribute__((ext_vector_type(2))) float v2f;
typedef __attribute__((ext_vector_type(8))) float v8f;

// ---------------------------------------------------------------------------
// K1: per-(target, relation) edge counts for mean aggregation (shared by both
// layers since the segment key tgt*R+rel is identical).
// ---------------------------------------------------------------------------
__global__ void count_kernel(const int* __restrict__ tgt,
                             const int* __restrict__ rel,
                             unsigned* __restrict__ cnt) {
    int e = blockIdx.x * blockDim.x + threadIdx.x;
    if (e >= EE) return;
    atomicAdd(&cnt[(size_t)tgt[e] * RR + rel[e]], 1u);
}

// ---------------------------------------------------------------------------
// K2: layer-1 message + mean aggregation.
//   msg1[e,h] = sum_b comp1[rel_e,b] * basis1[b, src_e, h]
//   hacc[tgt_e,h] += msg1[e,h] / max(cnt[tgt_e,rel_e],1)
// 4 threads per edge; each thread owns a float4 slice of H. basis1 rows are
// 64B contiguous; basis1 (192MB) ~= L2 so the gather streams from L2.
// ---------------------------------------------------------------------------
__global__ void layer1_kernel(const int* __restrict__ src,
                              const int* __restrict__ tgt,
                              const int* __restrict__ rel,
                              const float* __restrict__ basis1,
                              const float* __restrict__ comp1,
                              const unsigned* __restrict__ cnt,
                              float* __restrict__ hacc) {
    int t = blockIdx.x * blockDim.x + threadIdx.x;
    if (t >= EE * 4) return;
    int e = t >> 2;
    int q = t & 3;                       // which float4 of the H=16 row
    int s = src[e], d = tgt[e], r = rel[e];

    const float* bp = basis1 + (size_t)s * HH + q * 4;
    const float* cp = comp1 + (size_t)r * BB;

    float4 acc = make_float4(0.f, 0.f, 0.f, 0.f);
#pragma unroll
    for (int b = 0; b < BB; ++b) {
        float  w = cp[b];
        float4 v = *(const float4*)(bp + (size_t)b * NN * HH);
        acc.x = fmaf(w, v.x, acc.x);
        acc.y = fmaf(w, v.y, acc.y);
        acc.z = fmaf(w, v.z, acc.z);
        acc.w = fmaf(w, v.w, acc.w);
    }

    float c    = (float)cnt[(size_t)d * RR + r];
    float norm = 1.0f / fmaxf(c, 1.0f);
    float* op  = hacc + (size_t)d * HH + q * 4;
    atomicAdd(op + 0, acc.x * norm);
    atomicAdd(op + 1, acc.y * norm);
    atomicAdd(op + 2, acc.z * norm);
    atomicAdd(op + 3, acc.w * norm);
}

// ---------------------------------------------------------------------------
// K3: h = relu(agg + root1 + bias1)
// ---------------------------------------------------------------------------
__global__ void relu_kernel(float* __restrict__ h,
                            const float* __restrict__ root1,
                            const float* __restrict__ bias1) {
    int i = blockIdx.x * blockDim.x + threadIdx.x;
    if (i >= NN * HH) return;
    float v = h[i] + root1[i] + bias1[i & (HH - 1)];
    h[i] = fmaxf(v, 0.0f);
}

// ---------------------------------------------------------------------------
// K4: W2[r,h,c] = sum_b comp2[r,b] * basis2[b,h,c]   (tiny: 12800 outputs)
// ---------------------------------------------------------------------------
__global__ void w2_kernel(const float* __restrict__ comp2,
                          const float* __restrict__ basis2,
                          float* __restrict__ W2) {
    int i = blockIdx.x * blockDim.x + threadIdx.x;
    if (i >= RR * HH * CC) return;
    int r  = i / (HH * CC);
    int hc = i % (HH * CC);
    float a = 0.f;
#pragma unroll
    for (int b = 0; b < BB; ++b)
        a = fmaf(comp2[r * BB + b], basis2[b * HH * CC + hc], a);
    W2[i] = a;
}

// ---------------------------------------------------------------------------
// K5: layer-2 per-edge transform + mean aggregation into out.
//   msg2[e,c] = sum_h h[src_e,h] * W2[rel_e,h,c]
// 4 threads per edge, each owning 4 classes (float4 W2 loads; only 50
// relation matrices -> W2 stays resident in cache).
// ---------------------------------------------------------------------------
__global__ void layer2_kernel(const int* __restrict__ src,
                              const int* __restrict__ tgt,
                              const int* __restrict__ rel,
                              const float* __restrict__ h,
                              const float* __restrict__ W2,
                              const unsigned* __restrict__ cnt,
                              float* __restrict__ out) {
    int t = blockIdx.x * blockDim.x + threadIdx.x;
    if (t >= EE * 4) return;
    int e = t >> 2;
    int q = t & 3;                       // which float4 of C=16 classes
    int s = src[e], d = tgt[e], r = rel[e];

    const float* hp = h + (size_t)s * HH;
    const float* wp = W2 + (size_t)r * HH * CC + q * 4;

    float4 acc = make_float4(0.f, 0.f, 0.f, 0.f);
#pragma unroll
    for (int k = 0; k < HH; ++k) {
        float  hv = hp[k];
        float4 w  = *(const float4*)(wp + k * CC);
        acc.x = fmaf(hv, w.x, acc.x);
        acc.y = fmaf(hv, w.y, acc.y);
        acc.z = fmaf(hv, w.z, acc.z);
        acc.w = fmaf(hv, w.w, acc.w);
    }

    float c    = (float)cnt[(size_t)d * RR + r];
    float norm = 1.0f / fmaxf(c, 1.0f);
    float* op  = out + (size_t)d * CC + q * 4;
    atomicAdd(op + 0, acc.x * norm);
    atomicAdd(op + 1, acc.y * norm);
    atomicAdd(op + 2, acc.z * norm);
    atomicAdd(op + 3, acc.w * norm);
}

// ---------------------------------------------------------------------------
// K6: out += h @ root2 + bias2 via V_WMMA_F32_16X16X4_F32.
// One wave per 16-node tile; K=16 chained as 4 K=4 WMMAs (full f32).
// A layout (16x4 f32): M = lane&15; lanes 0-15 carry K={0,1}, lanes 16-31
// carry K={2,3} in the two A VGPRs. B layout (4x16): N = lane&15, same K
// split across lane halves. C VGPR v: M = v (+8 for upper lanes), N = lane&15.
// Exit guard is wave-uniform, so EXEC is all-ones for every executing WMMA.
// ---------------------------------------------------------------------------
__global__ void __launch_bounds__(256) root2_wmma_kernel(
        const float* __restrict__ h,
        const float* __restrict__ root2,
        const float* __restrict__ bias2,
        float* __restrict__ out) {
    int tile = blockIdx.x * 8 + (threadIdx.x >> 5);   // 8 waves per block
    if (tile >= NTILES) return;                        // wave-uniform exit
    int lane    = threadIdx.x & 31;
    int halfSel = lane >> 4;                           // 0: lanes 0-15, 1: 16-31
    int l       = lane & 15;
    int base    = tile * 16;

    v8f c = {};                                        // zero accumulator
#pragma unroll
    for (int kc = 0; kc < 4; ++kc) {
        int k0 = kc * 4 + halfSel * 2;
        v2f a, b;
        a.x = h[(size_t)(base + l) * HH + k0 + 0];
        a.y = h[(size_t)(base + l) * HH + k0 + 1];
        b.x = root2[(k0 + 0) * CC + l];
        b.y = root2[(k0 + 1) * CC + l];
        c = __builtin_amdgcn_wmma_f32_16x16x4_f32(
                /*neg_a=*/false, a, /*neg_b=*/false, b,
                /*c_mod=*/(short)0, c, /*reuse_a=*/false, /*reuse_b=*/false);
    }

    float bz    = bias2[l];
    int   mbase = halfSel * 8;
#pragma unroll
    for (int v = 0; v < 8; ++v) {
        size_t idx = (size_t)(base + mbase + v) * CC + l;
        out[idx] = out[idx] + c[v] + bz;               // after K5: no races
    }
}

// ---------------------------------------------------------------------------
// Workspace layout (needs ~26.5 MB):
//   [0)                cnt : uint32[N*R]   = 20,000,000 B
//   [20,000,256)       h   : float [N*H]   =  6,400,000 B
//   [26,400,512)       W2  : float [R*H*C] =     51,200 B
// ---------------------------------------------------------------------------
extern "C" void kernel_launch(void* const* d_in, const int* in_sizes, int n_in,
                              void* d_out, int out_size, void* d_ws, size_t ws_size,
                              hipStream_t stream) {
    const int*   ei     = (const int*)d_in[0];
    const int*   src    = ei;
    const int*   tgt    = ei + EE;
    const int*   rel    = (const int*)d_in[1];
    const float* basis1 = (const float*)d_in[2];
    const float* comp1  = (const float*)d_in[3];
    const float* root1  = (const float*)d_in[4];
    const float* bias1  = (const float*)d_in[5];
    const float* basis2 = (const float*)d_in[6];
    const float* comp2  = (const float*)d_in[7];
    const float* root2  = (const float*)d_in[8];
    const float* bias2  = (const float*)d_in[9];
    float*       out    = (float*)d_out;

    char*     ws  = (char*)d_ws;
    unsigned* cnt = (unsigned*)(ws);
    float*    h   = (float*)(ws + 20000256);
    float*    W2  = (float*)(ws + 26400512);

    hipMemsetAsync(cnt, 0, (size_t)NN * RR * sizeof(unsigned), stream);
    hipMemsetAsync(h,   0, (size_t)NN * HH * sizeof(float),    stream);
    hipMemsetAsync(out, 0, (size_t)NN * CC * sizeof(float),    stream);

    count_kernel <<<(EE        + 255) / 256, 256, 0, stream>>>(tgt, rel, cnt);
    layer1_kernel<<<(EE * 4    + 255) / 256, 256, 0, stream>>>(src, tgt, rel, basis1, comp1, cnt, h);
    relu_kernel  <<<(NN * HH   + 255) / 256, 256, 0, stream>>>(h, root1, bias1);
    w2_kernel    <<<(RR*HH*CC  + 255) / 256, 256, 0, stream>>>(comp2, basis2, W2);
    layer2_kernel<<<(EE * 4    + 255) / 256, 256, 0, stream>>>(src, tgt, rel, h, W2, cnt, out);
    root2_wmma_kernel<<<(NTILES + 7) / 8, 256, 0, stream>>>(h, root2, bias2, out);
}